// TriU_6347961664092
// MI455X (gfx1250) — compile-verified
//
#include <hip/hip_runtime.h>

typedef __attribute__((ext_vector_type(2))) float v2f;
typedef __attribute__((ext_vector_type(8))) float v8f;

#define TSTEP 32               // time steps (K and N of the GEMM)
#define ROWS_PER_TILE 128      // 8 waves x 16 rows per staged tile
#define TILES_PER_BLOCK 4      // rows per block = 512
#define LDS_STRIDE 36          // dwords: 16B-aligned rows, bank-conflict-free
#define TILE_F4 (ROWS_PER_TILE * TSTEP / 4)   // 1024 float4 per tile

// Out[B x 32] = X[B x 32] * tril(W)^T + b
// Per wave: one 16x32 output slab = two 16x16 f32 accumulators.
// K=32 swept with V_WMMA_F32_16X16X4_F32 (exact fp32 numerics):
//   c0 (t=0..15)  : K-steps 0..3 only (steps 4..7 are fully masked out)
//   c1 (t=16..31) : K-steps 0..7
// => 12 WMMAs per wave per tile.
__global__ void __launch_bounds__(256)
triu_fused_linears_wmma(const float* __restrict__ X,
                        const float* __restrict__ W,
                        const float* __restrict__ bias,
                        float* __restrict__ out)
{
    __shared__ float lds[2][ROWS_PER_TILE * LDS_STRIDE];

    const int tid  = threadIdx.x;
    const int lane = tid & 31;
    const int wave = tid >> 5;
    const size_t blockRow0 = (size_t)blockIdx.x * (ROWS_PER_TILE * TILES_PER_BLOCK);

    const int t_lo = lane & 15;
    const int hi   = lane >> 4;

    // ---- Build B fragments (tril(W)^T) once per block ----
    // Operand K convention (consistent between A and B):
    //   lanes 0-15 carry K = k0,k0+1 ; lanes 16-31 carry K = k0+2,k0+3.
    // Loads are unconditional (always in-bounds); mask applied via select
    // so no divergent branches are generated.
    v2f bf0[4];                       // tile0 columns t = 0..15, steps 0..3
    v2f bf1[8];                       // tile1 columns t = 16..31, steps 0..7
    const int t0 = t_lo;
    const int t1 = t_lo + 16;
#pragma unroll
    for (int i = 0; i < 4; ++i) {
        int k = i * 4 + 2 * hi;
        v2f w0 = *(const v2f*)(W + t0 * TSTEP + k);
        v2f w1 = *(const v2f*)(W + t1 * TSTEP + k);
        bf0[i][0] = (k     <= t0) ? w0[0] : 0.0f;   // v_cndmask
        bf0[i][1] = (k + 1 <= t0) ? w0[1] : 0.0f;
        bf1[i] = w1;                                 // k+1 <= 15 < 16 <= t1
    }
#pragma unroll
    for (int i = 4; i < 8; ++i) {
        int k = i * 4 + 2 * hi;
        v2f w1 = *(const v2f*)(W + t1 * TSTEP + k);
        bf1[i][0] = (k     <= t1) ? w1[0] : 0.0f;
        bf1[i][1] = (k + 1 <= t1) ? w1[1] : 0.0f;
    }
    const float bias0 = bias[t_lo];
    const float bias1 = bias[t_lo + 16];

    // ---- Ping-pong tile loop: stage tile t+1 while computing tile t ----
    const float4* __restrict__ Xv = (const float4*)(X + blockRow0 * TSTEP);
    const int waveRow0 = wave * 16;

    float4 stage[4];
#pragma unroll
    for (int i = 0; i < 4; ++i)
        stage[i] = Xv[i * 256 + tid];              // coalesced 128b loads
    {
        // store tile 0 into LDS buffer 0
#pragma unroll
        for (int i = 0; i < 4; ++i) {
            int idx = i * 256 + tid;
            int r = idx >> 3, c = idx & 7;
            *(float4*)&lds[0][r * LDS_STRIDE + c * 4] = stage[i];
        }
    }

    for (int t = 0; t < TILES_PER_BLOCK; ++t) {
        __syncthreads();                           // buf[t&1] fully staged

        // issue next tile's global loads first: overlap with WMMA/DS work
        if (t + 1 < TILES_PER_BLOCK) {
#pragma unroll
            for (int i = 0; i < 4; ++i)
                stage[i] = Xv[(t + 1) * TILE_F4 + i * 256 + tid];
        }

        // A fragments from LDS (ds_load_b64, conflict-free via stride 36)
        const float* aBase = &lds[t & 1][(waveRow0 + t_lo) * LDS_STRIDE];

        v8f c0 = {};
        v8f c1 = {};
#pragma unroll
        for (int i = 0; i < 4; ++i) {
            int k = i * 4 + 2 * hi;
            v2f a = *(const v2f*)(aBase + k);
            c0 = __builtin_amdgcn_wmma_f32_16x16x4_f32(
                     false, a, false, bf0[i], (short)0, c0, false, false);
            c1 = __builtin_amdgcn_wmma_f32_16x16x4_f32(
                     false, a, false, bf1[i], (short)0, c1, false, false);
        }
#pragma unroll
        for (int i = 4; i < 8; ++i) {              // tile0 fully masked here
            int k = i * 4 + 2 * hi;
            v2f a = *(const v2f*)(aBase + k);
            c1 = __builtin_amdgcn_wmma_f32_16x16x4_f32(
                     false, a, false, bf1[i], (short)0, c1, false, false);
        }

        // stage tile t+1 into the other buffer (other waves are reading
        // buf[t&1]; last reads of buf[(t+1)&1] were before this barrier)
        if (t + 1 < TILES_PER_BLOCK) {
#pragma unroll
            for (int i = 0; i < 4; ++i) {
                int idx = i * 256 + tid;
                int r = idx >> 3, c = idx & 7;
                *(float4*)&lds[(t + 1) & 1][r * LDS_STRIDE + c * 4] = stage[i];
            }
        }

        // ---- Store: C/D layout -> VGPR r holds rows (r, r+8) per lane half.
        // Lane half writes 16 consecutive floats per instruction (64B segs).
        float* outBase = out +
            (blockRow0 + (size_t)(t * ROWS_PER_TILE + waveRow0 + 8 * hi)) * TSTEP + t_lo;
#pragma unroll
        for (int r = 0; r < 8; ++r) {
            outBase[r * TSTEP]      = c0[r] + bias0;
            outBase[r * TSTEP + 16] = c1[r] + bias1;
        }
    }
}

extern "C" void kernel_launch(void* const* d_in, const int* in_sizes, int n_in,
                              void* d_out, int out_size, void* d_ws, size_t ws_size,
                              hipStream_t stream) {
    const float* X    = (const float*)d_in[0];  // [B, 32]
    const float* W    = (const float*)d_in[1];  // [32, 32] (lower-tri used)
    const float* bias = (const float*)d_in[2];  // [32]
    float* out        = (float*)d_out;          // [B, 32]

    const int rows   = in_sizes[0] / TSTEP;                      // B = 1048576
    const int blocks = rows / (ROWS_PER_TILE * TILES_PER_BLOCK); // 2048

    triu_fused_linears_wmma<<<blocks, 256, 0, stream>>>(X, W, bias, out);
}